// DistanceAwareGNN_1082331758596
// MI455X (gfx1250) — compile-verified
//
#include <hip/hip_runtime.h>
#include <hip/hip_bf16.h>

// Problem constants (compile-time, from reference)
#define B_      2
#define N_      512
#define M_TOT   (B_ * N_)   // 1024 rows
#define HID_    64
#define OUT_    128
#define LAYERS_ 3
#define HCAT_   256         // HID * (LAYERS+1)

typedef __attribute__((ext_vector_type(2))) float v2f;
typedef __attribute__((ext_vector_type(8))) float v8f;

// ---------------------------------------------------------------------------
// Generic f32 WMMA GEMM:  C[M,Nc] = [A1(:, :K1) | A2] @ W(K,Nc) (+bias)
//                         (+ rowscale[m]*rsvec[n]) (optional relu)
// One wave per 16x16 tile, V_WMMA_F32_16X16X4_F32, K stepped by 4.
// All M,Nc multiples of 16; K,K1 multiples of 4 (guaranteed by caller).
// ---------------------------------------------------------------------------
__global__ __launch_bounds__(256) void gemm_wmma_f32(
    const float* __restrict__ A1, int lda1, int K1,
    const float* __restrict__ A2, int lda2,
    const float* __restrict__ W,  int ldw,
    const float* __restrict__ bias,
    const float* __restrict__ rowscale, const float* __restrict__ rsvec,
    float* __restrict__ C, int ldc,
    int M, int Nc, int K, int do_relu)
{
    const int lane   = threadIdx.x & 31;
    const int wave   = threadIdx.x >> 5;
    const int tilesN = Nc >> 4;
    const int numTiles = (M >> 4) * tilesN;
    const int tile = blockIdx.x * 8 + wave;
    if (tile >= numTiles) return;              // wave-uniform; EXEC stays full

    const int tm   = (tile / tilesN) << 4;
    const int tn   = (tile % tilesN) << 4;
    const int half = lane >> 4;                // 0: K=k,k+1   1: K=k+2,k+3
    const int lp   = lane & 15;

    v8f acc = {};
    const int arow = tm + lp;

    for (int k = 0; k < K; k += 4) {
        const int ka = k + half * 2;
        v2f a, b;
        if (k < K1) {                          // uniform branch (K1 % 4 == 0)
            const float* ap = A1 + (size_t)arow * lda1 + ka;
            a.x = ap[0]; a.y = ap[1];
        } else {
            const float* ap = A2 + (size_t)arow * lda2 + (ka - K1);
            a.x = ap[0]; a.y = ap[1];
        }
        const float* wp = W + (size_t)ka * ldw + tn + lp;
        b.x = wp[0];
        b.y = wp[ldw];
        acc = __builtin_amdgcn_wmma_f32_16x16x4_f32(
            /*neg_a=*/false, a, /*neg_b=*/false, b,
            /*c_mod=*/(short)0, acc, /*reuse_a=*/false, /*reuse_b=*/false);
    }

    // C/D layout: lane<16 -> M=tm+r, N=tn+lane ; lane>=16 -> M=tm+8+r
    const int crow0 = tm + half * 8;
    const int cn    = tn + lp;
    const float bv = bias  ? bias[cn]  : 0.0f;
    const float rv = rsvec ? rsvec[cn] : 0.0f;
#pragma unroll
    for (int r = 0; r < 8; ++r) {
        const int crow = crow0 + r;
        float val = acc[r] + bv;
        if (rowscale) val = fmaf(rowscale[crow], rv, val);
        if (do_relu)  val = fmaxf(val, 0.0f);
        C[(size_t)crow * ldc + cn] = val;
    }
}

// ---------------------------------------------------------------------------
// Attention projections: p[m]=H[m,:D]@aw[:D], q[m]=H[m,:D]@aw[D:2D]
// ---------------------------------------------------------------------------
__global__ __launch_bounds__(256) void pq_kernel(
    const float* __restrict__ H, const float* __restrict__ attn_w,
    float* __restrict__ p, float* __restrict__ q, int D)
{
    const int m = blockIdx.x * 256 + threadIdx.x;
    const float* h = H + (size_t)m * HCAT_;
    float sp = 0.0f, sq = 0.0f;
    for (int k = 0; k < D; ++k) {
        const float hv = h[k];
        sp = fmaf(hv, attn_w[k],     sp);
        sq = fmaf(hv, attn_w[D + k], sq);
    }
    p[m] = sp; q[m] = sq;
}

// ---------------------------------------------------------------------------
// Fused per-row edge kernel. One block = one (b,i) row.
//   score_j = p_i + q_j + d_ij*awd + ab ; leaky_relu(0.2) ; mask(adj==0 -> -1e9)
//   attn = softmax_j ; a_j = attn_j * adj_ij ; asum = sum_j a_j
//   aggpre_i[d] = sum_j a_j * relu(u_i[d] + v_j[d] + d_ij*wd[d])   (b1 folded in u)
// ---------------------------------------------------------------------------
__global__ __launch_bounds__(256) void edge_kernel(
    const float* __restrict__ u,      // (1024,64)  = H@W_i + b1
    const float* __restrict__ v,      // (1024,64)  = H@W_j
    const float* __restrict__ p,      // (1024)
    const float* __restrict__ q,      // (1024)
    const float* __restrict__ adj,    // (1024,512)
    const float* __restrict__ dist,   // (1024,512)
    const float* __restrict__ msg_w1, // row 2D = wd
    const float* __restrict__ attn_w, // [2D] = awd scalar
    const float* __restrict__ attn_b, // [0]
    float* __restrict__ aggpre,       // (1024,64)
    float* __restrict__ asum,         // (1024)
    int D)
{
    const int row   = blockIdx.x;       // b*N + i
    const int t     = threadIdx.x;
    const int bbase = row & ~(N_ - 1);  // b*N

    __shared__ float s_a[N_];           // score -> exp -> a_j
    __shared__ float s_d[N_];
    __shared__ float s_adj[N_];
    __shared__ float s_r0[256];
    __shared__ float s_r1[256];

    const float awd = attn_w[2 * D];
    const float ab  = attn_b[0];
    const float pi  = p[row];

    // --- scores + row max ---
    float lmax = -3.402823466e38f;
    for (int j = t; j < N_; j += 256) {
        const float dij = dist[(size_t)row * N_ + j];
        const float aij = adj[(size_t)row * N_ + j];
        float s = fmaf(dij, awd, pi + q[bbase + j] + ab);
        s = (s > 0.0f) ? s : 0.2f * s;        // leaky relu
        if (aij == 0.0f) s = -1.0e9f;         // mask
        s_a[j] = s; s_d[j] = dij; s_adj[j] = aij;
        lmax = fmaxf(lmax, s);
    }
    s_r0[t] = lmax;
    __syncthreads();
    for (int off = 128; off > 0; off >>= 1) {
        if (t < off) s_r0[t] = fmaxf(s_r0[t], s_r0[t + off]);
        __syncthreads();
    }
    const float m = s_r0[0];
    __syncthreads();

    // --- exp + (sum e, sum e*adj) ---
    float lsum = 0.0f, lsuma = 0.0f;
    for (int j = t; j < N_; j += 256) {
        const float e = __expf(s_a[j] - m);
        s_a[j] = e;
        lsum  += e;
        lsuma += e * s_adj[j];
    }
    s_r0[t] = lsum; s_r1[t] = lsuma;
    __syncthreads();
    for (int off = 128; off > 0; off >>= 1) {
        if (t < off) { s_r0[t] += s_r0[t + off]; s_r1[t] += s_r1[t + off]; }
        __syncthreads();
    }
    const float inv = 1.0f / s_r0[0];
    if (t == 0) asum[row] = s_r1[0] * inv;
    for (int j = t; j < N_; j += 256)
        s_a[j] = s_a[j] * s_adj[j] * inv;     // a_j = attn_j * adj_ij
    __syncthreads();

    // --- weighted relu aggregation: 64 d-lanes x 4 j-groups ---
    const int d   = t & 63;
    const int grp = t >> 6;
    const float cd  = u[(size_t)row * HID_ + d];
    const float wdd = msg_w1[(size_t)(2 * D) * HID_ + d];
    const float* vb = v + (size_t)bbase * HID_;
    float acc = 0.0f;
    const int j0 = grp * (N_ / 4), j1 = j0 + (N_ / 4);
    for (int j = j0; j < j1; ++j) {
        const float aj = s_a[j];
        if (aj != 0.0f) {                     // uniform within wave: ~85% skipped
            float pre = fmaf(s_d[j], wdd, cd) + vb[(size_t)j * HID_ + d];
            pre = fmaxf(pre, 0.0f);
            acc = fmaf(aj, pre, acc);
        }
    }
    s_r0[t] = acc;
    __syncthreads();
    if (t < 64)
        aggpre[(size_t)row * HID_ + d] =
            s_r0[t] + s_r0[t + 64] + s_r0[t + 128] + s_r0[t + 192];
}

// ---------------------------------------------------------------------------
// Host orchestration.
// Input order (setup_inputs dict order, params in insertion order):
//  0 x, 1 adj, 2 dist, 3 in_w, 4 in_b,
//  per layer l (base=5+8l): msg_w1, msg_b1, msg_w2, msg_b2, attn_w, attn_b, upd_w, upd_b
//  29 out_w, 30 out_b
// ---------------------------------------------------------------------------
extern "C" void kernel_launch(void* const* d_in, const int* in_sizes, int n_in,
                              void* d_out, int out_size, void* d_ws, size_t ws_size,
                              hipStream_t stream)
{
    (void)in_sizes; (void)n_in; (void)out_size; (void)ws_size;

    const float* x     = (const float*)d_in[0];
    const float* adj   = (const float*)d_in[1];
    const float* dist  = (const float*)d_in[2];
    const float* in_w  = (const float*)d_in[3];
    const float* in_b  = (const float*)d_in[4];
    const float* out_w = (const float*)d_in[5 + 8 * LAYERS_];
    const float* out_b = (const float*)d_in[6 + 8 * LAYERS_];

    float* ws     = (float*)d_ws;
    float* H      = ws;                       // 1024 x 256
    float* u      = H      + (size_t)M_TOT * HCAT_;
    float* v      = u      + (size_t)M_TOT * HID_;
    float* aggpre = v      + (size_t)M_TOT * HID_;
    float* agg    = aggpre + (size_t)M_TOT * HID_;
    float* p      = agg    + (size_t)M_TOT * HID_;
    float* q      = p      + M_TOT;
    float* asum   = q      + M_TOT;

    auto gemm = [&](const float* A1, int lda1, int K1, const float* A2, int lda2,
                    const float* W, int ldw, const float* bias,
                    const float* rowscale, const float* rsvec,
                    float* C, int ldc, int M, int Nc, int K, int relu) {
        const int tiles  = (M / 16) * (Nc / 16);
        const int blocks = (tiles + 7) / 8;
        gemm_wmma_f32<<<dim3(blocks), dim3(256), 0, stream>>>(
            A1, lda1, K1, A2, lda2, W, ldw, bias, rowscale, rsvec,
            C, ldc, M, Nc, K, relu);
    };

    // h0 = x @ in_w + in_b  -> H[:, 0:64]
    gemm(x, HID_, HID_, x, HID_, in_w, HID_, in_b, nullptr, nullptr,
         H, HCAT_, M_TOT, HID_, HID_, 0);

    for (int l = 0; l < LAYERS_; ++l) {
        const int base = 5 + l * 8;
        const float* msg_w1 = (const float*)d_in[base + 0];
        const float* msg_b1 = (const float*)d_in[base + 1];
        const float* msg_w2 = (const float*)d_in[base + 2];
        const float* msg_b2 = (const float*)d_in[base + 3];
        const float* attn_w = (const float*)d_in[base + 4];
        const float* attn_b = (const float*)d_in[base + 5];
        const float* upd_w  = (const float*)d_in[base + 6];
        const float* upd_b  = (const float*)d_in[base + 7];
        const int D = HID_ * (l + 1);

        // u = H[:,:D] @ W_i + b1 ;  v = H[:,:D] @ W_j
        gemm(H, HCAT_, D, H, HCAT_, msg_w1, HID_, msg_b1, nullptr, nullptr,
             u, HID_, M_TOT, HID_, D, 0);
        gemm(H, HCAT_, D, H, HCAT_, msg_w1 + (size_t)D * HID_, HID_, nullptr,
             nullptr, nullptr, v, HID_, M_TOT, HID_, D, 0);

        pq_kernel<<<dim3(M_TOT / 256), dim3(256), 0, stream>>>(H, attn_w, p, q, D);

        edge_kernel<<<dim3(M_TOT), dim3(256), 0, stream>>>(
            u, v, p, q, adj, dist, msg_w1, attn_w, attn_b, aggpre, asum, D);

        // agg = aggpre @ msg_w2 + asum * msg_b2
        gemm(aggpre, HID_, HID_, aggpre, HID_, msg_w2, HID_, nullptr,
             asum, msg_b2, agg, HID_, M_TOT, HID_, HID_, 0);

        // h_{l+1} = relu([H[:,:D] | agg] @ upd_w + upd_b) -> H[:, 64*(l+1) : 64*(l+2)]
        gemm(H, HCAT_, D, agg, HID_, upd_w, HID_, upd_b, nullptr, nullptr,
             H + (size_t)HID_ * (l + 1), HCAT_, M_TOT, HID_, D + HID_, 1);
    }

    // out = H @ out_w + out_b
    gemm(H, HCAT_, HCAT_, H, HCAT_, out_w, OUT_, out_b, nullptr, nullptr,
         (float*)d_out, OUT_, M_TOT, OUT_, HCAT_, 0);
}